// Generator_15470472200220
// MI455X (gfx1250) — compile-verified
//
#include <hip/hip_runtime.h>

typedef __attribute__((ext_vector_type(2))) float v2f;
typedef __attribute__((ext_vector_type(8))) float v8f;

#define B_  64
#define ZD_ 512
#define E_  512
#define L_  8192
#define NL_ 1000

#define WMMA_F32(a, b, c) \
  __builtin_amdgcn_wmma_f32_16x16x4_f32(false, (a), false, (b), (short)0, (c), false, false)

// ---------------- ye0[b,e] = W_yemb[e, y[b]] + b_yemb[e] ----------------
__global__ void gather_yemb(const float* __restrict__ Wy,
                            const float* __restrict__ bias,
                            const int* __restrict__ y,
                            float* __restrict__ out) {
  int idx = blockIdx.x * blockDim.x + threadIdx.x;   // B_*E_ threads
  int b = idx >> 9;          // /512
  int e = idx & 511;
  out[idx] = Wy[e * NL_ + y[b]] + bias[e];
}

// ---------------- out = x + x @ W^T + bias ; x:[64,512], W:[512,512] ----
// Split-K x4: each 16x16 tile is owned by 4 waves (K chunks of 128),
// partials reduced through LDS. Block = 8 waves = 2 tiles (same M-tile).
// Grid: 64 blocks = 4 m_tiles x 16 n-pairs  -> 512 waves per layer.
__global__ void irm_layer(const float* __restrict__ x,
                          const float* __restrict__ W,
                          const float* __restrict__ bias,
                          float* __restrict__ out) {
  __shared__ float part[2][4][256];                  // [tile][k-split][m*16+n]

  const int lane = threadIdx.x & 31;
  const int wave = threadIdx.x >> 5;
  const int t    = wave >> 2;                        // tile within block: 0/1
  const int kw   = wave & 3;                         // k-split: 0..3
  const int m_tile = blockIdx.x >> 4;                // 0..3
  const int m0     = m_tile * 16;
  const int n0base = (blockIdx.x & 15) * 32;         // 2 tiles of 16
  const int half = lane >> 4, lr = lane & 15;
  const int k0 = kw * 128;

  // A fragment: row m0+lr, K pair at 2*half (ISA 32-bit A 16x4 layout)
  const float* ap = x + (m0 + lr) * E_ + k0 + 2 * half;
  // B fragment: out[.,n]=sum_k x[.,k]*W[n,k] -> row n of W, same K pair
  const float* bp = W + (n0base + t * 16 + lr) * E_ + k0 + 2 * half;

  v8f acc = {};
#pragma unroll 8
  for (int k = 0; k < 128; k += 4) {
    v2f a = *(const v2f*)(ap + k);
    v2f b = *(const v2f*)(bp + k);
    acc = WMMA_F32(a, b, acc);
  }

  // Dump partial tile: element (r,lane) -> M = r+8*half, N = lr
#pragma unroll
  for (int r = 0; r < 8; ++r)
    part[t][kw][(r + 8 * half) * 16 + lr] = acc[r];

  __syncthreads();

  // Reduce 4 partials + residual + bias; 512 outputs over 256 threads.
#pragma unroll
  for (int i = threadIdx.x; i < 512; i += 256) {
    int tt  = i >> 8;
    int idx = i & 255;
    float sum = part[tt][0][idx] + part[tt][1][idx] +
                part[tt][2][idx] + part[tt][3][idx];
    int m = m0 + (idx >> 4);
    int n = n0base + tt * 16 + (idx & 15);
    out[m * E_ + n] = sum + x[m * E_ + n] + bias[n];
  }
}

// ---------------- s[b] = sum_e ye[b,e] * wproj[e] -----------------------
__global__ void sscal_kernel(const float* __restrict__ ye,
                             const float* __restrict__ wproj,
                             float* __restrict__ s) {
  int b = blockIdx.x;
  int lane = threadIdx.x;                 // 32 threads = 1 wave
  float sum = 0.f;
  for (int k = lane; k < E_; k += 32) sum += ye[b * E_ + k] * wproj[k];
#pragma unroll
  for (int off = 16; off > 0; off >>= 1) sum += __shfl_xor(sum, off, 32);
  if (lane == 0) s[b] = sum;
}

// ---------------- out[b,l] = (zz[b,:] . Wz[l,:]) * s[b] -----------------
// [64,512] x [512,8192] f32 GEMM, fused per-row scaling epilogue.
// Each wave computes 2 N-tiles (reuses the A fragment: 3 loads / 2 WMMA).
// Grid: 128 blocks x 8 waves = 1024 waves; block shares one M-tile in L1.
__global__ void zlat_kernel(const float* __restrict__ zz,
                            const float* __restrict__ Wz,
                            const float* __restrict__ s,
                            float* __restrict__ out) {
  const int lane = threadIdx.x & 31;
  const int wave = threadIdx.x >> 5;
  const int m_tile = blockIdx.x >> 5;                 // 0..3
  const int pair   = (blockIdx.x & 31) * 8 + wave;    // 0..255
  const int m0 = m_tile * 16;
  const int n0 = pair * 32;                           // two 16-wide tiles
  const int half = lane >> 4, lr = lane & 15;

  const float* ap  = zz + (m0 + lr) * ZD_ + 2 * half;
  const float* bp0 = Wz + (n0 + lr) * ZD_ + 2 * half;
  const float* bp1 = Wz + (n0 + 16 + lr) * ZD_ + 2 * half;

  v8f acc0 = {}, acc1 = {};
#pragma unroll 8
  for (int k = 0; k < ZD_; k += 4) {
    v2f a  = *(const v2f*)(ap  + k);
    v2f b0 = *(const v2f*)(bp0 + k);
    v2f b1 = *(const v2f*)(bp1 + k);
    acc0 = WMMA_F32(a, b0, acc0);
    acc1 = WMMA_F32(a, b1, acc1);
  }

#pragma unroll
  for (int r = 0; r < 8; ++r) {
    int m = m0 + r + 8 * half;
    float sm = s[m];
    out[m * L_ + n0 + lr]      = acc0[r] * sm;
    out[m * L_ + n0 + 16 + lr] = acc1[r] * sm;
  }
}

extern "C" void kernel_launch(void* const* d_in, const int* in_sizes, int n_in,
                              void* d_out, int out_size, void* d_ws, size_t ws_size,
                              hipStream_t stream) {
  const float* z      = (const float*)d_in[0];
  const int*   y      = (const int*)  d_in[1];
  const float* W_yemb = (const float*)d_in[2];
  const float* b_yemb = (const float*)d_in[3];
  const float* Wy1    = (const float*)d_in[4];
  const float* by1    = (const float*)d_in[5];
  const float* Wy2    = (const float*)d_in[6];
  const float* by2    = (const float*)d_in[7];
  const float* Wz1    = (const float*)d_in[8];
  const float* bz1    = (const float*)d_in[9];
  const float* Wz2    = (const float*)d_in[10];
  const float* bz2    = (const float*)d_in[11];
  const float* W_zlat = (const float*)d_in[12];
  const float* w_proj = (const float*)d_in[13];
  float* out = (float*)d_out;

  float* ws  = (float*)d_ws;
  float* ye0 = ws;                    // 64*512
  float* ye1 = ye0 + B_ * E_;
  float* ye2 = ye1 + B_ * E_;
  float* zz1 = ye2 + B_ * E_;
  float* zz2 = zz1 + B_ * ZD_;
  float* sc  = zz2 + B_ * ZD_;        // 64

  gather_yemb<<<(B_ * E_) / 256, 256, 0, stream>>>(W_yemb, b_yemb, y, ye0);
  irm_layer<<<64, 256, 0, stream>>>(ye0, Wy1, by1, ye1);
  irm_layer<<<64, 256, 0, stream>>>(ye1, Wy2, by2, ye2);
  irm_layer<<<64, 256, 0, stream>>>(z,   Wz1, bz1, zz1);
  irm_layer<<<64, 256, 0, stream>>>(zz1, Wz2, bz2, zz2);
  sscal_kernel<<<B_, 32, 0, stream>>>(ye2, w_proj, sc);
  zlat_kernel<<<128, 256, 0, stream>>>(zz2, W_zlat, sc, out);
}